// SynapticSNN_6519760355814
// MI455X (gfx1250) — compile-verified
//
#include <hip/hip_runtime.h>

typedef __attribute__((ext_vector_type(2))) float v2f;
typedef __attribute__((ext_vector_type(8))) float v8f;

#define B_    4096
#define NI_   256
#define NH_   2048
#define NO_   128
#define STEPS 100
#define ALPHA_ 0.9f

// ---------------------------------------------------------------------------
// Kernel 1: fused  cur1 = x @ W1^T + b1  ->  100-step Synaptic recurrence
//           -> final spikes written to ws (f32 0/1, exact)
// One wave computes one 16x16 tile of cur1 via V_WMMA_F32_16X16X4_F32,
// then runs the elementwise recurrence on its 8 accumulator registers.
// Fragment layout (16x16x4 f32):
//   A: lane (khi*16+m) vgpr j  holds A[m][khi*2+j]  -> contiguous float2 in x row
//   B: lane (khi*16+n) vgpr j  holds B[khi*2+j][n]  -> contiguous float2 in W1 row
//   C/D: vgpr r, lanes 0-15 -> (row0+r, col0+lane); lanes 16-31 -> (row0+r+8, ...)
// ---------------------------------------------------------------------------
__global__ __launch_bounds__(256) void snn_gemm1_recur(
    const float* __restrict__ x,    // [B_, NI_]
    const float* __restrict__ W1,   // [NH_, NI_]
    const float* __restrict__ b1,   // [NH_]
    const float* __restrict__ betap,
    float* __restrict__ spk)        // [B_, NH_] workspace
{
    const int wave = threadIdx.x >> 5;
    const int lane = threadIdx.x & 31;
    const int tile = blockIdx.x * 8 + wave;
    const int tilesN = NH_ / 16;                 // 128
    const int tm = tile / tilesN;
    const int tn = tile - tm * tilesN;
    const int row0 = tm << 4;
    const int col0 = tn << 4;
    const int mn  = lane & 15;
    const int khi = lane >> 4;

    const float* ap = x  + (size_t)(row0 + mn) * NI_ + (khi << 1);
    const float* bp = W1 + (size_t)(col0 + mn) * NI_ + (khi << 1);

    v8f acc = {};
#pragma unroll 8
    for (int k = 0; k < NI_; k += 4) {
        v2f a = *(const v2f*)(ap + k);
        v2f b = *(const v2f*)(bp + k);
        // (neg_a, A, neg_b, B, c_mod, C, reuse_a, reuse_b)
        acc = __builtin_amdgcn_wmma_f32_16x16x4_f32(
            false, a, false, b, (short)0, acc, false, false);
    }

    float beta = *betap;
    beta = fminf(fmaxf(beta, 0.0f), 1.0f);
    const float bias = b1[col0 + mn];            // column-indexed: same for all 8 rows

    float c[8], syn[8], mem[8];
#pragma unroll
    for (int i = 0; i < 8; ++i) {
        c[i] = acc[i] + bias;
        syn[i] = 0.0f;
        mem[i] = 0.0f;
    }

    // reset-by-subtraction recurrence; only the final spike matters.
#pragma unroll 1
    for (int t = 0; t < STEPS; ++t) {
#pragma unroll
        for (int i = 0; i < 8; ++i) {
            const float reset = (mem[i] > 1.0f) ? 1.0f : 0.0f;  // pre-update mem
            syn[i] = fmaf(ALPHA_, syn[i], c[i]);
            mem[i] = fmaf(beta, mem[i], syn[i]) - reset;        // THRESHOLD = 1.0
        }
    }

    float* op = spk + (size_t)(row0 + khi * 8) * NH_ + (col0 + mn);
#pragma unroll
    for (int i = 0; i < 8; ++i)
        op[(size_t)i * NH_] = ((mem[i] - 1.0f) > 0.0f) ? 1.0f : 0.0f;
}

// ---------------------------------------------------------------------------
// Kernel 2: cur2 = spk @ W2^T + b2   via V_WMMA_F32_16X16X4_F32, K = 2048.
// spk (32 MB) and W2 (1 MB) are L2-resident on MI455X (192 MB L2).
// ---------------------------------------------------------------------------
__global__ __launch_bounds__(256) void snn_gemm2(
    const float* __restrict__ spk,  // [B_, NH_]
    const float* __restrict__ W2,   // [NO_, NH_]
    const float* __restrict__ b2,   // [NO_]
    float* __restrict__ out)        // [B_, NO_]
{
    const int wave = threadIdx.x >> 5;
    const int lane = threadIdx.x & 31;
    const int tile = blockIdx.x * 8 + wave;
    const int tilesN = NO_ / 16;                 // 8
    const int tm = tile / tilesN;
    const int tn = tile - tm * tilesN;
    const int row0 = tm << 4;
    const int col0 = tn << 4;
    const int mn  = lane & 15;
    const int khi = lane >> 4;

    const float* ap = spk + (size_t)(row0 + mn) * NH_ + (khi << 1);
    const float* bp = W2  + (size_t)(col0 + mn) * NH_ + (khi << 1);

    v8f acc = {};
#pragma unroll 8
    for (int k = 0; k < NH_; k += 4) {
        v2f a = *(const v2f*)(ap + k);
        v2f b = *(const v2f*)(bp + k);
        acc = __builtin_amdgcn_wmma_f32_16x16x4_f32(
            false, a, false, b, (short)0, acc, false, false);
    }

    const float bias = b2[col0 + mn];
    float* op = out + (size_t)(row0 + khi * 8) * NO_ + (col0 + mn);
#pragma unroll
    for (int i = 0; i < 8; ++i)
        op[(size_t)i * NO_] = acc[i] + bias;
}

// ---------------------------------------------------------------------------
extern "C" void kernel_launch(void* const* d_in, const int* in_sizes, int n_in,
                              void* d_out, int out_size, void* d_ws, size_t ws_size,
                              hipStream_t stream) {
    (void)in_sizes; (void)n_in; (void)out_size; (void)ws_size;
    const float* x    = (const float*)d_in[0];   // [4096, 256]
    const float* W1   = (const float*)d_in[1];   // [2048, 256]
    const float* b1   = (const float*)d_in[2];   // [2048]
    const float* W2   = (const float*)d_in[3];   // [128, 2048]
    const float* b2   = (const float*)d_in[4];   // [128]
    const float* beta = (const float*)d_in[5];   // scalar

    float* spk = (float*)d_ws;                   // [4096, 2048] = 32 MB
    float* out = (float*)d_out;                  // [4096, 128]

    const int tiles1 = (B_ / 16) * (NH_ / 16);   // 32768 tiles, 8 waves/block
    snn_gemm1_recur<<<tiles1 / 8, 256, 0, stream>>>(x, W1, b1, beta, spk);

    const int tiles2 = (B_ / 16) * (NO_ / 16);   // 2048 tiles
    snn_gemm2<<<tiles2 / 8, 256, 0, stream>>>(spk, W2, b2, out);
}